// causalAttention_v2_76278619177478
// MI455X (gfx1250) — compile-verified
//
#include <hip/hip_runtime.h>

typedef float v2f __attribute__((ext_vector_type(2)));
typedef float v8f __attribute__((ext_vector_type(8)));

#define S_TOK 8192
#define DIN   512
#define DOUT  64
#define NBLK  (S_TOK / 16)   // 512 row blocks

union F8 { v8f v; float f[8]; };

__device__ __forceinline__ v8f wmma4(v2f a, v2f b, v8f c) {
  // D = A(16x4 f32) * B(4x16 f32) + C(16x16 f32)
  return __builtin_amdgcn_wmma_f32_16x16x4_f32(
      /*neg_a=*/false, a, /*neg_b=*/false, b,
      /*c_mod=*/(short)0, c, /*reuse_a=*/false, /*reuse_b=*/false);
}

__device__ __forceinline__ float rowmax16(float x) {
  x = fmaxf(x, __shfl_xor(x, 1, 32));
  x = fmaxf(x, __shfl_xor(x, 2, 32));
  x = fmaxf(x, __shfl_xor(x, 4, 32));
  x = fmaxf(x, __shfl_xor(x, 8, 32));
  return x;
}
__device__ __forceinline__ float rowsum16(float x) {
  x += __shfl_xor(x, 1, 32);
  x += __shfl_xor(x, 2, 32);
  x += __shfl_xor(x, 4, 32);
  x += __shfl_xor(x, 8, 32);
  return x;
}

// ---------------------------------------------------------------------------
// Kernel 1: Q/K/V projection.  y = x @ W^T  (x:[S,512], W:[64,512], y:[S,64])
// One wave computes one 16x16 output tile. K=512 reduction split into two
// independent 64-deep WMMA chains for matrix-pipe ILP.
// grid = (512 row-blocks, 12 = {Q,K,V} x 4 col-tiles), block = 32 (one wave).
// Q is pre-scaled by 1/sqrt(64).
// ---------------------------------------------------------------------------
__global__ __launch_bounds__(32)
void qkv_proj_kernel(const float* __restrict__ x,
                     const float* __restrict__ Wq,
                     const float* __restrict__ Wk,
                     const float* __restrict__ Wv,
                     float* __restrict__ qkv /* [3][S][64] */) {
  const int lane = threadIdx.x & 31;
  const int hi   = lane >> 4;     // which half-wave
  const int ln   = lane & 15;
  const int iblk = blockIdx.x;       // row block
  const int mat  = blockIdx.y >> 2;  // 0=Q 1=K 2=V
  const int nt   = blockIdx.y & 3;   // output column tile

  const float* W = (mat == 0) ? Wq : (mat == 1 ? Wk : Wv);
  const float* arow = x + (size_t)(iblk * 16 + ln) * DIN;  // A: x row (M = ln)
  const float* brow = W + (size_t)(nt * 16 + ln) * DIN;    // B: W row (N = ln)

  F8 acc0 = {}, acc1 = {};        // two independent accumulation chains
#pragma unroll 4
  for (int k0 = 0; k0 < DIN; k0 += 8) {
    // 16x4 f32 A layout: lanes0-15 hold K=k0,k0+1 ; lanes16-31 hold K=k0+2,k0+3
    v2f a0 = *(const v2f*)(arow + k0 + 2 * hi);
    v2f b0 = *(const v2f*)(brow + k0 + 2 * hi);
    v2f a1 = *(const v2f*)(arow + k0 + 4 + 2 * hi);
    v2f b1 = *(const v2f*)(brow + k0 + 4 + 2 * hi);
    acc0.v = wmma4(a0, b0, acc0.v);
    acc1.v = wmma4(a1, b1, acc1.v);
  }
  F8 acc; acc.v = acc0.v + acc1.v;

  const float scl = (mat == 0) ? 0.125f : 1.0f;  // fold 1/sqrt(d_k) into Q
  float* out = qkv + (size_t)mat * S_TOK * DOUT;
#pragma unroll
  for (int v = 0; v < 8; ++v) {
    const int row = iblk * 16 + v + 8 * hi;      // C/D layout: M = v + 8*hi
    out[(size_t)row * DOUT + nt * 16 + ln] = acc.f[v] * scl;
  }
}

// ---------------------------------------------------------------------------
// Kernel 2: causal flash attention, one wave per 16 query rows.
// block = 64 (2 waves), grid = 256. Wave 0 owns i-block b, wave 1 owns
// 511-b: every block executes exactly 513 j-iterations total -> zero load
// imbalance while spreading 512 waves over 256 WGPs.
// ---------------------------------------------------------------------------
__global__ __launch_bounds__(64)
void flash_attn_kernel(const float* __restrict__ Q,
                       const float* __restrict__ K,
                       const float* __restrict__ V,
                       float* __restrict__ out) {
  __shared__ float plds[2 * 256];   // 1KB P-tile per wave (layout transpose)

  const int lane = threadIdx.x & 31;
  const int wid  = threadIdx.x >> 5;
  const int hi   = lane >> 4;
  const int ln   = lane & 15;
  const int iblk = wid ? (NBLK - 1 - blockIdx.x) : blockIdx.x;  // 0..511
  float* pt = &plds[wid * 256];

  // Q A-fragments for full K=64 reduction: 16 chunks of K=4 (32 VGPRs)
  v2f qa[16];
  const float* qrow = Q + (size_t)(iblk * 16 + ln) * DOUT;
#pragma unroll
  for (int c = 0; c < 16; ++c)
    qa[c] = *(const v2f*)(qrow + 4 * c + 2 * hi);

  F8 o[4] = {};                    // O accumulator, 4 x (16x16) C/D tiles
  float m[8], l[8];                // per-row running max / sum (row = v + 8*hi)
#pragma unroll
  for (int v = 0; v < 8; ++v) { m[v] = -__builtin_inff(); l[v] = 0.0f; }

  for (int j = 0; j <= iblk; ++j) {
    if (j < iblk) {  // prefetch next K/V tile into cache (global_prefetch_b8)
      __builtin_prefetch(K + (size_t)((j + 1) * 16 + ln) * DOUT, 0, 3);
      __builtin_prefetch(V + (size_t)((j + 1) * 16 + ln) * DOUT, 0, 3);
    }

    // ---- S = (Q/sqrt(dk)) @ K_j^T : two independent 8-deep WMMA chains ----
    F8 s0 = {}, s1 = {};
    const float* krow = K + (size_t)(j * 16 + ln) * DOUT;  // B: N = ln (K row)
#pragma unroll
    for (int c = 0; c < 8; ++c) {
      v2f kb0 = *(const v2f*)(krow + 4 * c + 2 * hi);
      v2f kb1 = *(const v2f*)(krow + 4 * (c + 8) + 2 * hi);
      s0.v = wmma4(qa[c], kb0, s0.v);
      s1.v = wmma4(qa[c + 8], kb1, s1.v);
    }
    F8 s; s.v = s0.v + s1.v;

    // ---- causal mask + online softmax, per row v+8*hi ----
    const bool diag = (j == iblk);
#pragma unroll
    for (int v = 0; v < 8; ++v) {
      float sv = s.f[v];
      if (diag && ln > v + 8 * hi) sv = -__builtin_inff();  // n > m -> masked
      const float mn = fmaxf(m[v], rowmax16(sv));
      const float sc = __expf(m[v] - mn);
      const float p  = __expf(sv - mn);
      l[v] = l[v] * sc + rowsum16(p);
      m[v] = mn;
#pragma unroll
      for (int t = 0; t < 4; ++t) o[t].f[v] *= sc;   // rescale O rows
      s.f[v] = p;
    }

    // ---- transpose P: C/D layout -> A layout, via per-wave LDS tile ----
#pragma unroll
    for (int v = 0; v < 8; ++v)
      pt[(v + 8 * hi) * 16 + ln] = s.f[v];
    __builtin_amdgcn_wave_barrier();
    asm volatile("s_wait_dscnt 0" ::: "memory");   // wave-local LDS fence

    // ---- O += P @ V_j : 4 chunks (K=4) x 4 independent d-tile chains ----
    const float* vbase = V + (size_t)(j * 16) * DOUT;
#pragma unroll
    for (int c = 0; c < 4; ++c) {
      v2f pa = *(const v2f*)(pt + ln * 16 + 4 * c + 2 * hi);  // A-frag of P
      const float* v0 = vbase + (size_t)(4 * c + 2 * hi) * DOUT + ln;
#pragma unroll
      for (int t = 0; t < 4; ++t) {
        v2f vb;                       // B-frag of V: rows 4c+2hi{,+1}, col 16t+ln
        vb.x = v0[16 * t];
        vb.y = v0[16 * t + DOUT];
        o[t].v = wmma4(pa, vb, o[t].v);
      }
    }
  }

  // ---- normalize and write O ----
#pragma unroll
  for (int v = 0; v < 8; ++v) {
    const float inv = 1.0f / l[v];
    const int row = iblk * 16 + v + 8 * hi;
#pragma unroll
    for (int t = 0; t < 4; ++t)
      out[(size_t)row * DOUT + 16 * t + ln] = o[t].f[v] * inv;
  }
}

// ---------------------------------------------------------------------------
extern "C" void kernel_launch(void* const* d_in, const int* in_sizes, int n_in,
                              void* d_out, int out_size, void* d_ws, size_t ws_size,
                              hipStream_t stream) {
  const float* x  = (const float*)d_in[0];
  const float* Wq = (const float*)d_in[1];
  const float* Wk = (const float*)d_in[2];
  const float* Wv = (const float*)d_in[3];
  float* qkv = (float*)d_ws;              // Q | K | V, each S*64 f32 (6 MB)
  float* out = (float*)d_out;

  dim3 gA(NBLK, 12);
  qkv_proj_kernel<<<gA, 32, 0, stream>>>(x, Wq, Wk, Wv, qkv);

  const float* Qp = qkv;
  const float* Kp = qkv + (size_t)S_TOK * DOUT;
  const float* Vp = qkv + (size_t)2 * S_TOK * DOUT;
  flash_attn_kernel<<<256, 64, 0, stream>>>(Qp, Kp, Vp, out);
}